// ComplexMixture_71313636983193
// MI455X (gfx1250) — compile-verified
//
#include <hip/hip_runtime.h>
#include <stdint.h>

// CDNA5 / gfx1250: wave32, WMMA f32 16x16x4 + async global->LDS staging.
typedef float v2f __attribute__((ext_vector_type(2)));
typedef float v8f __attribute__((ext_vector_type(8)));

#define BATCH 32
#define SEQ   128
#define NDIM  256

#define KC       16          // K (seq) chunk staged per double-buffer phase
#define NCHUNK   (SEQ / KC)  // 8
#define LDSTRIDE 72          // floats per staged row (64 data + 8 pad, bank-friendly)

// smem layout: [buf 0/1][panel: A_r,A_i,B_r,B_i][KC * LDSTRIDE]
#define PANEL_ELTS (KC * LDSTRIDE)

// Each wave issues 2 async b128 copies per chunk; 16 waves x 2 = 32 issues
// cover 4 panels x 8 slots (each b128 issue moves 2 rows of 64 floats).
__device__ __forceinline__ void issue_chunk(float* smem_pb,
                                            const float* Rb, const float* Ib,
                                            int n0, int m0, int c,
                                            int wave, int lane) {
#pragma unroll
    for (int j = 0; j < 2; ++j) {
        const int t    = wave * 2 + j;
        const int arr  = t >> 3;                 // 0:A_r 1:A_i 2:B_r 3:B_i
        const int slot = t & 7;
        const int row  = 2 * slot + (lane >> 4); // row within chunk (0..15)
        const int col4 = (lane & 15) * 4;        // 4-float (16B) column slice

        const float* src  = (arr & 1) ? Ib : Rb;
        const int    colb = (arr & 2) ? m0 : n0;
        const float* g = src + (size_t)(c * KC + row) * NDIM + colb + col4;

        float* l = smem_pb + arr * PANEL_ELTS + row * LDSTRIDE + col4;
        // addrspace(3) offset lives in the low 32 bits of the generic pointer
        const uint32_t lds_off = (uint32_t)(uintptr_t)l;

        // GLOBAL_LOAD_ASYNC_TO_LDS_B128: vdst = LDS byte address, vaddr = 64b
        // global address, GV mode. Tracked with ASYNCcnt.
        asm volatile("global_load_async_to_lds_b128 %0, %1, off"
                     :: "v"(lds_off), "v"(g) : "memory");
    }
}

__global__ __launch_bounds__(512) void complex_mixture_wmma_f32(
    const float* __restrict__ real,    // [B,S,N]
    const float* __restrict__ imag,    // [B,S,N]
    const float* __restrict__ weight,  // [B,S]
    float*       __restrict__ out)     // out_r [B,N,N] then out_i [B,N,N]
{
    __shared__ float smem[2][4][PANEL_ELTS];   // 36,864 B
    __shared__ float w_lds[SEQ];               // 512 B

    const int b    = blockIdx.z;
    const int n0   = blockIdx.y * 64;
    const int m0   = blockIdx.x * 64;
    const int wave = threadIdx.x >> 5;
    const int lane = threadIdx.x & 31;
    const int half = lane >> 4;   // K-pair selector within 4-wide K chunk
    const int l    = lane & 15;   // M index for A, N index for B

    const int wn = wave >> 2;     // wave's 16-row slice of the 64x64 tile
    const int wm = wave & 3;      // wave's 16-col slice
    const int nloc = wn * 16 + l; // column into staged A panels
    const int mloc = wm * 16 + l; // column into staged B panels

    const float* Rb = real + (size_t)b * SEQ * NDIM;
    const float* Ib = imag + (size_t)b * SEQ * NDIM;

    if (threadIdx.x < SEQ) w_lds[threadIdx.x] = weight[b * SEQ + threadIdx.x];

    // Prologue: stage chunk 0 into buffer 0.
    issue_chunk(&smem[0][0][0], Rb, Ib, n0, m0, 0, wave, lane);

    v8f acc_r = {0.f, 0.f, 0.f, 0.f, 0.f, 0.f, 0.f, 0.f};
    v8f acc_i = {0.f, 0.f, 0.f, 0.f, 0.f, 0.f, 0.f, 0.f};

#pragma unroll 2
    for (int c = 0; c < NCHUNK; ++c) {
        const int p = c & 1;

        if (c + 1 < NCHUNK) {
            issue_chunk(&smem[p ^ 1][0][0], Rb, Ib, n0, m0, c + 1, wave, lane);
            // Async loads complete in order: <=2 outstanding => chunk c landed.
            asm volatile("s_wait_asynccnt 2" ::: "memory");
        } else {
            asm volatile("s_wait_asynccnt 0" ::: "memory");
        }
        __syncthreads();   // all waves' chunk-c writes visible

        const float* sa_r = &smem[p][0][0];
        const float* sa_i = &smem[p][1][0];
        const float* sb_r = &smem[p][2][0];
        const float* sb_i = &smem[p][3][0];

#pragma unroll
        for (int kk = 0; kk < KC / 4; ++kk) {
            const int sl = 4 * kk + 2 * half;   // row pair within chunk
            const int sg = c * KC + sl;         // global s for the weight
            const float w0 = w_lds[sg];
            const float w1 = w_lds[sg + 1];

            // WMMA f32 16x16x4 operand layout (ISA 7.12.2):
            //   lanes 0-15: VGPR0=K0, VGPR1=K1 ; lanes 16-31: VGPR0=K2, VGPR1=K3
            v2f ar = { sa_r[sl * LDSTRIDE + nloc], sa_r[(sl + 1) * LDSTRIDE + nloc] };
            v2f ai = { sa_i[sl * LDSTRIDE + nloc], sa_i[(sl + 1) * LDSTRIDE + nloc] };
            v2f br = { sb_r[sl * LDSTRIDE + mloc], sb_r[(sl + 1) * LDSTRIDE + mloc] };
            v2f bi = { sb_i[sl * LDSTRIDE + mloc], sb_i[(sl + 1) * LDSTRIDE + mloc] };

            // Pre-scale A by per-s weight; explicit negation for -wr*i
            // (f32 WMMA requires NEG[1:0] == 0 per ISA).
            v2f awr  = {  w0 * ar.x,  w1 * ar.y };
            v2f awi  = {  w0 * ai.x,  w1 * ai.y };
            v2f awrn = { -awr.x, -awr.y };

            // out_r += (w r)^T r + (w i)^T i
            acc_r = __builtin_amdgcn_wmma_f32_16x16x4_f32(false, awr,  false, br,
                                                          (short)0, acc_r, false, false);
            acc_r = __builtin_amdgcn_wmma_f32_16x16x4_f32(false, awi,  false, bi,
                                                          (short)0, acc_r, false, false);
            // out_i += (w i)^T r - (w r)^T i
            acc_i = __builtin_amdgcn_wmma_f32_16x16x4_f32(false, awi,  false, br,
                                                          (short)0, acc_i, false, false);
            acc_i = __builtin_amdgcn_wmma_f32_16x16x4_f32(false, awrn, false, bi,
                                                          (short)0, acc_i, false, false);
        }
        __syncthreads();   // reads of buffer p done before it is re-staged
    }

    // C/D layout: VGPR j, lanes 0-15 -> M=j ; lanes 16-31 -> M=j+8.
    float* outr = out + (size_t)b * NDIM * NDIM;
    float* outi = out + (size_t)BATCH * NDIM * NDIM + (size_t)b * NDIM * NDIM;
    const int col = m0 + wm * 16 + l;
#pragma unroll
    for (int j = 0; j < 8; ++j) {
        const int row = n0 + wn * 16 + j + 8 * half;
        outr[(size_t)row * NDIM + col] = acc_r[j];
        outi[(size_t)row * NDIM + col] = acc_i[j];
    }
}

extern "C" void kernel_launch(void* const* d_in, const int* in_sizes, int n_in,
                              void* d_out, int out_size, void* d_ws, size_t ws_size,
                              hipStream_t stream) {
    const float* real   = (const float*)d_in[0];
    const float* imag   = (const float*)d_in[1];
    const float* weight = (const float*)d_in[2];
    float* out = (float*)d_out;

    dim3 grid(NDIM / 64, NDIM / 64, BATCH);  // (4, 4, 32)
    dim3 block(512);                         // 16 waves of 32
    hipLaunchKernelGGL(complex_mixture_wmma_f32, grid, block, 0, stream,
                       real, imag, weight, out);
}